// LiquidEchoModel_75892072121033
// MI455X (gfx1250) — compile-verified
//
#include <hip/hip_runtime.h>
#include <math.h>

// ---- model constants (shapes fixed by the reference: D=128, N=256, LUT=4096) ----
#define LUT_N     4096
#define LUT_MASK  4095
#define D_DIM     128
#define N_DIM     256
#define TWO_D     256
#define PHI_F     1.6180339887498949f
#define LUT_SCALE 651.8986469044033f      // LUT_N / (2*pi)
#define INV_SCALE_D 0.08838834764831845f  // 1/sqrt(128)

typedef float v2f __attribute__((ext_vector_type(2)));
typedef float v8f __attribute__((ext_vector_type(8)));

__device__ __forceinline__ int lut_idx(float th) {
    // mod(floor(th*scale), 4096): & works for negatives on two's complement
    return ((int)floorf(th * LUT_SCALE)) & LUT_MASK;
}

// =====================================================================
// Phase 1: sequential recurrence. One block per batch element.
// 1024 threads: theta phase uses all of them (n = tid>>2, d-quarter = tid&3),
// per-d phases use threads 0..127. LUT tables live in LDS.
// Writes X[(b*S+s)*256 + k] : k<128 -> xr, k>=128 -> xi.
// =====================================================================
__global__ __launch_bounds__(1024)
void recurrence_kernel(const int* __restrict__ ids, const float* __restrict__ t,
                       const float* __restrict__ emb,
                       const float* __restrict__ ic_w, const float* __restrict__ ic_b,
                       const float* __restrict__ Wres, const float* __restrict__ Bres,
                       const float* __restrict__ attn_cos, const float* __restrict__ attn_sin,
                       const float* __restrict__ opr_w, const float* __restrict__ opi_w,
                       const float* __restrict__ wq, const float* __restrict__ bq,
                       const float* __restrict__ wo, const float* __restrict__ bo,
                       const float* __restrict__ gate_scales,
                       const float* __restrict__ sin_t, const float* __restrict__ cos_t,
                       float* __restrict__ X, int S, int L)
{
    __shared__ float s_sin[LUT_N];
    __shared__ float s_cos[LUT_N];
    __shared__ float s_xr[D_DIM], s_xi[D_DIM], s_xc[D_DIM];
    __shared__ float s_cs[N_DIM], s_ss[N_DIM];
    __shared__ float s_resr[D_DIM], s_resi[D_DIM];
    __shared__ float s_memr[2][D_DIM], s_memi[2][D_DIM];
    __shared__ float s_red[D_DIM];

    const int tid = threadIdx.x;
    const int b   = blockIdx.x;

    for (int i = tid; i < LUT_N; i += blockDim.x) { s_sin[i] = sin_t[i]; s_cos[i] = cos_t[i]; }
    if (tid < D_DIM) {
        s_memr[0][tid] = 0.f; s_memi[0][tid] = 0.f;
        s_memr[1][tid] = 0.f; s_memi[1][tid] = 0.f;
    }
    __syncthreads();

    for (int s = 0; s < S; ++s) {
        const int   tok = ids[b * S + s];
        const float ts  = t[s];
        if (tid < D_DIM) {
            s_xr[tid] = emb[(size_t)tok * TWO_D + tid];
            s_xi[tid] = emb[(size_t)tok * TWO_D + D_DIM + tid];
        }
        __syncthreads();

        for (int l = 0; l < L; ++l) {
            // xc = concat(xr,xi) @ ic_w[l].T + ic_b[l]
            if (tid < D_DIM) {
                const float* wrow = ic_w + ((size_t)l * D_DIM + tid) * TWO_D;
                float acc = ic_b[l * D_DIM + tid];
                #pragma unroll 4
                for (int k = 0; k < D_DIM; ++k) acc += s_xr[k] * wrow[k];
                #pragma unroll 4
                for (int k = 0; k < D_DIM; ++k) acc += s_xi[k] * wrow[D_DIM + k];
                s_xc[tid] = acc;
            }
            __syncthreads();

            // theta = xc/(1+|Wres|) + Bres + ts ; LUT sin/cos ; reduce over D
            {
                const int n  = tid >> 2;
                const int dq = tid & 3;
                const size_t base = ((size_t)l * N_DIM + n) * D_DIM;
                float cs = 0.f, ss = 0.f;
                #pragma unroll 4
                for (int i = 0; i < 32; ++i) {
                    const int d = dq + 4 * i;            // quad-contiguous 16B loads
                    const float w  = 1.0f + fabsf(Wres[base + d]);
                    const float th = s_xc[d] / w + Bres[base + d] + ts;
                    const int idx = lut_idx(th);
                    cs += s_cos[idx] * attn_cos[base + d];
                    ss += s_sin[idx] * attn_sin[base + d];
                }
                cs += __shfl_xor(cs, 1, 32); cs += __shfl_xor(cs, 2, 32);
                ss += __shfl_xor(ss, 1, 32); ss += __shfl_xor(ss, 2, 32);
                if (dq == 0) { s_cs[n] = cs; s_ss[n] = ss; }
            }
            __syncthreads();

            // res_r/res_i = silu(sum @ op*_w.T) ; interference contributions
            if (tid < D_DIM) {
                const float* rrow = opr_w + ((size_t)l * D_DIM + tid) * N_DIM;
                const float* irow = opi_w + ((size_t)l * D_DIM + tid) * N_DIM;
                float ar = 0.f, ai = 0.f;
                #pragma unroll 4
                for (int n = 0; n < N_DIM; ++n) { ar += s_cs[n] * rrow[n]; ai += s_ss[n] * irow[n]; }
                s_resr[tid] = ar / (1.0f + expf(-ar));
                s_resi[tid] = ai / (1.0f + expf(-ai));
                const float wqv = 1.0f + fabsf(wq[l * D_DIM + tid]);
                const float thq = s_xr[tid] / wqv + bq[l * D_DIM + tid];
                const int idx = lut_idx(thq);
                s_red[tid] = s_cos[idx] * s_xr[tid] + s_sin[idx] * s_xi[tid];
            }
            __syncthreads();
            for (int off = D_DIM / 2; off > 0; off >>= 1) {
                if (tid < off) s_red[tid] += s_red[tid + off];
                __syncthreads();
            }
            const float inter = s_red[0];
            const float alpha = expf(-0.5f * (1.0f - tanhf(inter * INV_SCALE_D)));  // K_SENS=1

            // memory blend + echo gate + state update
            if (tid < D_DIM) {
                const float mr = s_memr[l][tid], mi = s_memi[l][tid];
                const float br = alpha * s_xr[tid] + (1.0f - alpha) * mr;
                const float bi = alpha * s_xi[tid] + (1.0f - alpha) * mi;
                const float wlo  = 1.0f + fabsf(wo[l * D_DIM + tid]);
                const float bov  = bo[l * D_DIM + tid];
                const float tphi = ts * PHI_F;
                const int i1 = lut_idx(br / wlo + bov + tphi);
                const int i2 = lut_idx(bi / wlo + bov + tphi);
                const float sor = s_sin[i1], cor = s_cos[i1];
                const float soi = s_sin[i2], coi = s_cos[i2];
                const float er = cor * coi - sor * soi;
                const float ei = cor * soi + sor * coi;
                s_memr[l][tid] = er; s_memi[l][tid] = ei;
                const float gs = gate_scales[l];
                s_xr[tid] = s_resr[tid] * (1.0f + gs * er) + s_xr[tid];
                s_xi[tid] = s_resi[tid] * (1.0f + gs * ei) + s_xi[tid];
            }
            __syncthreads();
        }

        if (tid < D_DIM) {
            const size_t m = (size_t)b * S + s;
            X[m * TWO_D + tid]         = s_xr[tid];
            X[m * TWO_D + D_DIM + tid] = s_xi[tid];
        }
        __syncthreads();
    }
}

// =====================================================================
// Phase 2: logits GEMM with V_WMMA_F32_16X16X4_F32.
// C(1024 x V) = X(1024x256) @ [out_real_w ; out_imag_w]^T (K=256).
// One wave computes a 16(M) x 64(N) strip: 4 accumulators share each
// A fragment. A frag: lane(M=lane&15) holds K pair (2*(lane>>4)+{0,1}) as
// float2; B frag: lane(N=lane&15) holds the same K pair from W's row.
// Output stores are NON-TEMPORAL: 206 MB written once, never re-read —
// streaming it past L2 keeps the 51 MB weight matrix resident in the
// 192 MB L2 across all 64 M-tiles.
// =====================================================================
__global__ __launch_bounds__(256)
void logits_kernel(const float* __restrict__ X,
                   const float* __restrict__ Wr, const float* __restrict__ Wi,
                   float* __restrict__ out, int V, int NT64, int totalWaves)
{
    const int w = blockIdx.x * 8 + (threadIdx.x >> 5);
    if (w >= totalWaves) return;            // wave-uniform: EXEC stays all-ones
    const int lane  = threadIdx.x & 31;
    const int mt    = w / NT64;
    const int nt    = w % NT64;
    const int mBase = mt * 16;
    const int nBase = nt * 64;
    const int laneM = lane & 15;            // also laneN for B/C fragments
    const int kh    = lane >> 4;            // K half-select

    const float* arow = X + (size_t)(mBase + laneM) * TWO_D;
    const int v0 = nBase + 0  + laneM, c0 = (v0 < V) ? v0 : (V - 1);
    const int v1 = nBase + 16 + laneM, c1 = (v1 < V) ? v1 : (V - 1);
    const int v2 = nBase + 32 + laneM, c2 = (v2 < V) ? v2 : (V - 1);
    const int v3 = nBase + 48 + laneM, c3 = (v3 < V) ? v3 : (V - 1);

    v8f acc0 = {}, acc1 = {}, acc2 = {}, acc3 = {};

    // K = 0..127 : real part
    {
        const float* r0 = Wr + (size_t)c0 * D_DIM;
        const float* r1 = Wr + (size_t)c1 * D_DIM;
        const float* r2 = Wr + (size_t)c2 * D_DIM;
        const float* r3 = Wr + (size_t)c3 * D_DIM;
        #pragma unroll 4
        for (int kk = 0; kk < 32; ++kk) {
            const int kc = kk * 4 + 2 * kh;
            const v2f a  = *(const v2f*)(arow + kc);
            const v2f b0 = *(const v2f*)(r0 + kc);
            const v2f b1 = *(const v2f*)(r1 + kc);
            const v2f b2 = *(const v2f*)(r2 + kc);
            const v2f b3 = *(const v2f*)(r3 + kc);
            acc0 = __builtin_amdgcn_wmma_f32_16x16x4_f32(false, a, false, b0, (short)0, acc0, false, false);
            acc1 = __builtin_amdgcn_wmma_f32_16x16x4_f32(false, a, false, b1, (short)0, acc1, false, false);
            acc2 = __builtin_amdgcn_wmma_f32_16x16x4_f32(false, a, false, b2, (short)0, acc2, false, false);
            acc3 = __builtin_amdgcn_wmma_f32_16x16x4_f32(false, a, false, b3, (short)0, acc3, false, false);
        }
    }
    // K = 128..255 : imag part
    {
        const float* i0 = Wi + (size_t)c0 * D_DIM;
        const float* i1 = Wi + (size_t)c1 * D_DIM;
        const float* i2 = Wi + (size_t)c2 * D_DIM;
        const float* i3 = Wi + (size_t)c3 * D_DIM;
        #pragma unroll 4
        for (int kk = 0; kk < 32; ++kk) {
            const int kc = kk * 4 + 2 * kh;
            const v2f a  = *(const v2f*)(arow + D_DIM + kc);
            const v2f b0 = *(const v2f*)(i0 + kc);
            const v2f b1 = *(const v2f*)(i1 + kc);
            const v2f b2 = *(const v2f*)(i2 + kc);
            const v2f b3 = *(const v2f*)(i3 + kc);
            acc0 = __builtin_amdgcn_wmma_f32_16x16x4_f32(false, a, false, b0, (short)0, acc0, false, false);
            acc1 = __builtin_amdgcn_wmma_f32_16x16x4_f32(false, a, false, b1, (short)0, acc1, false, false);
            acc2 = __builtin_amdgcn_wmma_f32_16x16x4_f32(false, a, false, b2, (short)0, acc2, false, false);
            acc3 = __builtin_amdgcn_wmma_f32_16x16x4_f32(false, a, false, b3, (short)0, acc3, false, false);
        }
    }

    // C/D layout: element (vgpr j, lane): M = j + 8*(lane>>4), N = lane&15
    // Non-temporal: stream logits to HBM, don't evict L2-resident weights.
    #pragma unroll
    for (int j = 0; j < 8; ++j) {
        const int m = mBase + j + 8 * kh;
        const size_t rowOff = (size_t)m * V;
        if (v0 < V) __builtin_nontemporal_store(acc0[j], out + rowOff + v0);
        if (v1 < V) __builtin_nontemporal_store(acc1[j], out + rowOff + v1);
        if (v2 < V) __builtin_nontemporal_store(acc2[j], out + rowOff + v2);
        if (v3 < V) __builtin_nontemporal_store(acc3[j], out + rowOff + v3);
    }
}

// =====================================================================
extern "C" void kernel_launch(void* const* d_in, const int* in_sizes, int n_in,
                              void* d_out, int out_size, void* d_ws, size_t ws_size,
                              hipStream_t stream)
{
    const int*   ids       = (const int*)  d_in[0];
    const float* t         = (const float*)d_in[1];
    const float* emb       = (const float*)d_in[2];
    const float* ic_w      = (const float*)d_in[3];
    const float* ic_b      = (const float*)d_in[4];
    const float* Wres      = (const float*)d_in[5];
    const float* Bres      = (const float*)d_in[6];
    const float* attn_cos  = (const float*)d_in[7];
    const float* attn_sin  = (const float*)d_in[8];
    const float* opr_w     = (const float*)d_in[9];
    const float* opi_w     = (const float*)d_in[10];
    const float* wq        = (const float*)d_in[11];
    const float* bq        = (const float*)d_in[12];
    const float* wo        = (const float*)d_in[13];
    const float* bo        = (const float*)d_in[14];
    const float* gates     = (const float*)d_in[15];
    const float* out_r_w   = (const float*)d_in[16];
    const float* out_i_w   = (const float*)d_in[17];
    const float* sin_t     = (const float*)d_in[18];
    const float* cos_t     = (const float*)d_in[19];

    const int S  = in_sizes[1];            // 64
    const int BS = in_sizes[0];            // B*S = 1024
    const int B  = BS / S;                 // 16
    const int L  = in_sizes[15];           // 2
    const int D  = in_sizes[11] / L;       // 128
    const int V  = in_sizes[16] / D;       // 50257
    (void)D;

    float* X = (float*)d_ws;               // (B*S) x 2D fp32 state matrix (1 MB)

    recurrence_kernel<<<B, 1024, 0, stream>>>(ids, t, emb, ic_w, ic_b, Wres, Bres,
                                              attn_cos, attn_sin, opr_w, opi_w,
                                              wq, bq, wo, bo, gates, sin_t, cos_t,
                                              X, S, L);

    const int MT    = BS / 16;             // 64 M-tiles
    const int NT64  = (V + 63) / 64;       // 786 N-supertiles
    const int waves = MT * NT64;
    const int blocks = (waves + 7) / 8;    // 8 waves per 256-thread block
    logits_kernel<<<blocks, 256, 0, stream>>>(X, out_r_w, out_i_w, (float*)d_out,
                                              V, NT64, waves);
}